// MSHGEncoderLayer_72834055406086
// MI455X (gfx1250) — compile-verified
//
#include <hip/hip_runtime.h>
#include <math.h>

#define N_NODES 50000
#define N_EDGES 500000
#define R_REL   3
#define IN_DIM  256
#define D_HEAD  32
#define H_HEADS 8
#define HD      256     // H*D
#define RIN_DIM 64
#define RELW    512     // H*2*D == H*RHID
#define SLOPE   0.2f
#define EPW     4       // edges per wave in scatter kernel (E % (8*EPW) == 0)

typedef float v2f __attribute__((ext_vector_type(2)));
typedef float v8f __attribute__((ext_vector_type(8)));

// ---------------------------------------------------------------------------
// Order-preserving float<->uint encoding for atomicMax-based segment max.
// ---------------------------------------------------------------------------
__device__ __forceinline__ unsigned enc_f32(float f) {
  unsigned u = __float_as_uint(f);
  return (u & 0x80000000u) ? ~u : (u | 0x80000000u);
}
__device__ __forceinline__ float dec_f32(unsigned u) {
  return __uint_as_float((u & 0x80000000u) ? (u & 0x7FFFFFFFu) : ~u);
}
__device__ __forceinline__ float lrelu(float x) {
  return x > 0.f ? x : SLOPE * x;
}

// ---------------------------------------------------------------------------
// K0: tiny GEMVs.  a[r] = rel_emb[r] @ W_rel[r]   (H x 2D, flat 512)
//                  rel_out[r] = rel_emb[r] @ W_prop[r] + b_prop[r]
// ---------------------------------------------------------------------------
__global__ __launch_bounds__(512)
void k_rel_small(const float* __restrict__ rel_emb, const float* __restrict__ W_rel,
                 const float* __restrict__ W_prop, const float* __restrict__ b_prop,
                 float* __restrict__ a_out, float* __restrict__ rel_out) {
  const int r = blockIdx.x;
  const int t = threadIdx.x;                      // 0..511
  const float* re = rel_emb + r * RIN_DIM;
  const float* wr = W_rel  + (size_t)r * RIN_DIM * RELW + t;
  const float* wp = W_prop + (size_t)r * RIN_DIM * RELW + t;
  float acc_a = 0.f, acc_p = 0.f;
#pragma unroll 8
  for (int i = 0; i < RIN_DIM; ++i) {
    const float rei = re[i];
    acc_a += rei * wr[i * RELW];
    acc_p += rei * wp[i * RELW];
  }
  a_out[r * RELW + t]   = acc_a;
  rel_out[r * RELW + t] = acc_p + b_prop[r * RELW + t];
}

// ---------------------------------------------------------------------------
// K1: feat = h @ W   (50000x256 @ 256x256), fp32 WMMA 16x16x4.
// Block = 256 threads = 8 waves. Block tile 64(M) x 64(N), K-step 32.
// Wave w: row-tile rt = w>>1, col-tiles 2*(w&1), 2*(w&1)+1  -> 2 accumulators.
// A fragment (16x4 f32): lane m=lane&15, kh=lane>>4 holds {A[m][2kh],A[m][2kh+1]}
// B fragment (4x16 f32): lane holds {B[2kh][n], B[2kh+1][n]}, n = lane&15
//   -> B staged TRANSPOSED in LDS (BsT[col][k], stride 34) so the fragment is
//      a contiguous, 8B-aligned float2 -> single ds_load_b64, no packing movs.
// C/D (16x16 f32): VGPR j = row j + 8*kh, lane n = column.
// ---------------------------------------------------------------------------
__global__ __launch_bounds__(256)
void k_gemm_feat(const float* __restrict__ h, const float* __restrict__ W,
                 float* __restrict__ feat) {
  __shared__ float As [64][34];   // [row][k]  pad 34: conflict-free frag reads
  __shared__ float BsT[64][34];   // [col][k]  transposed B tile, same padding

  const int tid  = threadIdx.x;
  const int lane = tid & 31;
  const int wave = tid >> 5;
  const int m    = lane & 15;      // row (A) / column (B,C)
  const int kh   = lane >> 4;      // K-half selector
  const int rt   = wave >> 1;      // 0..3 row tile
  const int ct0  = (wave & 1) * 2; // first of two col tiles

  const int row0 = blockIdx.x * 64;
  const int col0 = blockIdx.y * 64;

  v8f acc0 = {}; v8f acc1 = {};

  for (int k0 = 0; k0 < IN_DIM; k0 += 32) {
    // stage A: 64 rows x 32 k  (512 float4s / 256 threads)
#pragma unroll
    for (int it = 0; it < 2; ++it) {
      const int idx = tid + it * 256;
      const int ar = idx >> 3;            // 0..63
      const int ak = (idx & 7) * 4;       // 0,4,...,28
      int gr = row0 + ar; if (gr >= N_NODES) gr = N_NODES - 1;  // clamp, EXEC stays full
      const float4 v = *(const float4*)(h + (size_t)gr * IN_DIM + k0 + ak);
      As[ar][ak + 0] = v.x; As[ar][ak + 1] = v.y;
      As[ar][ak + 2] = v.z; As[ar][ak + 3] = v.w;
    }
    // stage B transposed: 32 k x 64 cols -> BsT[col][k]
#pragma unroll
    for (int it = 0; it < 2; ++it) {
      const int idx = tid + it * 256;
      const int br = idx >> 4;            // k    0..31
      const int bc = (idx & 15) * 4;      // col  0..60
      const float4 v = *(const float4*)(W + (size_t)(k0 + br) * HD + col0 + bc);
      BsT[bc + 0][br] = v.x; BsT[bc + 1][br] = v.y;
      BsT[bc + 2][br] = v.z; BsT[bc + 3][br] = v.w;
    }
    __syncthreads();

#pragma unroll
    for (int kk = 0; kk < 32; kk += 4) {
      const v2f afrag = *(const v2f*)&As [rt * 16 + m]        [kk + 2 * kh]; // 8B aligned
      const v2f b0    = *(const v2f*)&BsT[ct0 * 16 + m]       [kk + 2 * kh];
      const v2f b1    = *(const v2f*)&BsT[(ct0 + 1) * 16 + m] [kk + 2 * kh];
      acc0 = __builtin_amdgcn_wmma_f32_16x16x4_f32(false, afrag, false, b0,
                                                   (short)0, acc0, false, false);
      acc1 = __builtin_amdgcn_wmma_f32_16x16x4_f32(false, afrag, false, b1,
                                                   (short)0, acc1, false, false);
    }
    __syncthreads();
  }

#pragma unroll
  for (int j = 0; j < 8; ++j) {
    const int row = row0 + rt * 16 + j + 8 * kh;
    if (row < N_NODES) {
      feat[(size_t)row * HD + col0 + ct0 * 16 + m]       = acc0[j];
      feat[(size_t)row * HD + col0 + (ct0 + 1) * 16 + m] = acc1[j];
    }
  }
}

// ---------------------------------------------------------------------------
// K2: attention logits  e_l[n,h] = feat[n,h,:] . a_l[h],  e_r likewise.
// ---------------------------------------------------------------------------
__global__ __launch_bounds__(256)
void k_attn_logits(const float* __restrict__ feat, const float* __restrict__ a,
                   float* __restrict__ e_l, float* __restrict__ e_r) {
  const int t = blockIdx.x * 256 + threadIdx.x;
  if (t >= N_NODES * H_HEADS) return;
  const int n = t >> 3, hh = t & 7;
  const float* f  = feat + (size_t)n * HD + hh * D_HEAD;
  const float* al = a + hh * 64;
  const float* ar = al + D_HEAD;
  float sl = 0.f, sr = 0.f;
#pragma unroll
  for (int d = 0; d < D_HEAD; ++d) {
    const float x = f[d];
    sl += x * al[d];
    sr += x * ar[d];
  }
  e_l[t] = sl; e_r[t] = sr;
}

// ---------------------------------------------------------------------------
// K3: segment max over dst via native u32 atomicMax on order-preserving keys.
// ---------------------------------------------------------------------------
__global__ __launch_bounds__(256)
void k_edge_max(const int* __restrict__ src, const int* __restrict__ dst,
                const float* __restrict__ e_l, const float* __restrict__ e_r,
                unsigned* __restrict__ m_enc) {
  const int t = blockIdx.x * 256 + threadIdx.x;
  if (t >= N_EDGES * H_HEADS) return;
  const int eidx = t >> 3, hh = t & 7;
  const int s = src[eidx], d = dst[eidx];
  const float e = lrelu(e_l[s * H_HEADS + hh] + e_r[d * H_HEADS + hh]);
  atomicMax(&m_enc[d * H_HEADS + hh], enc_f32(e));
}

// ---------------------------------------------------------------------------
// K4: segment sum of exp(e - m[dst])  (recomputes e: cheaper than 16MB spill)
// ---------------------------------------------------------------------------
__global__ __launch_bounds__(256)
void k_edge_sum(const int* __restrict__ src, const int* __restrict__ dst,
                const float* __restrict__ e_l, const float* __restrict__ e_r,
                const unsigned* __restrict__ m_enc, float* __restrict__ s_sum) {
  const int t = blockIdx.x * 256 + threadIdx.x;
  if (t >= N_EDGES * H_HEADS) return;
  const int eidx = t >> 3, hh = t & 7;
  const int s = src[eidx], d = dst[eidx];
  const float e  = lrelu(e_l[s * H_HEADS + hh] + e_r[d * H_HEADS + hh]);
  const float mm = dec_f32(m_enc[d * H_HEADS + hh]);
  unsafeAtomicAdd(&s_sum[d * H_HEADS + hh], __expf(e - mm));
}

// ---------------------------------------------------------------------------
// K5: message scatter  out[dst] += alpha * feat[src].
// One wave per EPW(=4) edges; lane handles 8 contiguous columns (head=lane>>2).
// All gathers for the 4 edges are issued before any compute/atomics -> 8
// outstanding global_load_b128 per wave for latency hiding; atomics are
// fire-and-forget native global_atomic_add_f32 resolving in L2.
// E % (8 waves * EPW) == 0, so no tail predication: EXEC stays full.
// ---------------------------------------------------------------------------
__global__ __launch_bounds__(256)
void k_edge_scatter(const int* __restrict__ src, const int* __restrict__ dst,
                    const float* __restrict__ e_l, const float* __restrict__ e_r,
                    const unsigned* __restrict__ m_enc, const float* __restrict__ s_sum,
                    const float* __restrict__ feat, float* __restrict__ out) {
  const int wave = threadIdx.x >> 5;
  const int lane = threadIdx.x & 31;
  const int base = (blockIdx.x * 8 + wave) * EPW;
  const int hh   = lane >> 2;                      // 8 cols/lane -> head = lane/4

  int sIdx[EPW], dIdx[EPW];
#pragma unroll
  for (int u = 0; u < EPW; ++u) {
    sIdx[u] = src[base + u];
    dIdx[u] = dst[base + u];
  }

  float4 f0[EPW], f1[EPW];
#pragma unroll
  for (int u = 0; u < EPW; ++u) {
    const float4* fs = (const float4*)(feat + (size_t)sIdx[u] * HD + lane * 8);
    f0[u] = fs[0];
    f1[u] = fs[1];
  }

#pragma unroll
  for (int u = 0; u < EPW; ++u) {
    const int s = sIdx[u], d = dIdx[u];
    const float e  = lrelu(e_l[s * H_HEADS + hh] + e_r[d * H_HEADS + hh]);
    const float mm = dec_f32(m_enc[d * H_HEADS + hh]);
    const float ss = s_sum[d * H_HEADS + hh];
    const float alpha = __expf(e - mm) / (ss + 1e-9f);

    float* op = out + (size_t)d * HD + lane * 8;
    unsafeAtomicAdd(op + 0, alpha * f0[u].x);
    unsafeAtomicAdd(op + 1, alpha * f0[u].y);
    unsafeAtomicAdd(op + 2, alpha * f0[u].z);
    unsafeAtomicAdd(op + 3, alpha * f0[u].w);
    unsafeAtomicAdd(op + 4, alpha * f1[u].x);
    unsafeAtomicAdd(op + 5, alpha * f1[u].y);
    unsafeAtomicAdd(op + 6, alpha * f1[u].z);
    unsafeAtomicAdd(op + 7, alpha * f1[u].w);
  }
}

// ---------------------------------------------------------------------------
// K6: final ReLU over out_feats (vectorized float4)
// ---------------------------------------------------------------------------
__global__ __launch_bounds__(256)
void k_relu(float* __restrict__ x, int n4) {
  const int t = blockIdx.x * 256 + threadIdx.x;
  if (t >= n4) return;
  float4 v = ((float4*)x)[t];
  v.x = fmaxf(v.x, 0.f); v.y = fmaxf(v.y, 0.f);
  v.z = fmaxf(v.z, 0.f); v.w = fmaxf(v.w, 0.f);
  ((float4*)x)[t] = v;
}

// ---------------------------------------------------------------------------
extern "C" void kernel_launch(void* const* d_in, const int* in_sizes, int n_in,
                              void* d_out, int out_size, void* d_ws, size_t ws_size,
                              hipStream_t stream) {
  const float* h         = (const float*)d_in[0];
  const float* rel_emb   = (const float*)d_in[1];
  const float* W_src     = (const float*)d_in[2];
  const float* W_rel     = (const float*)d_in[3];
  const float* W_prop    = (const float*)d_in[4];
  const float* b_prop    = (const float*)d_in[5];
  const int*   edges_src = (const int*)d_in[6];
  const int*   edges_dst = (const int*)d_in[7];

  float* out       = (float*)d_out;
  float* out_feats = out;                                        // R*N*HD
  float* dst_feats = out + (size_t)R_REL * N_NODES * HD;         // R*N*HD
  float* rel_out   = out + (size_t)2 * R_REL * N_NODES * HD;     // R*512

  // workspace layout (~6.4 MB)
  float*    a_all = (float*)d_ws;                                // R*512
  float*    e_l   = a_all + R_REL * RELW;                        // N*H
  float*    e_r   = e_l + (size_t)N_NODES * H_HEADS;             // N*H
  unsigned* m_enc = (unsigned*)(e_r + (size_t)N_NODES * H_HEADS);// N*H
  float*    s_sum = (float*)(m_enc + (size_t)N_NODES * H_HEADS); // N*H

  const size_t nh = (size_t)N_NODES * H_HEADS;

  // zero the atomically-accumulated output region
  hipMemsetAsync(out_feats, 0, (size_t)R_REL * N_NODES * HD * sizeof(float), stream);

  k_rel_small<<<R_REL, 512, 0, stream>>>(rel_emb, W_rel, W_prop, b_prop, a_all, rel_out);

  for (int r = 0; r < R_REL; ++r) {
    const float* W    = W_src + (size_t)r * IN_DIM * HD;
    float*       feat = dst_feats + (size_t)r * N_NODES * HD;
    float*       outr = out_feats + (size_t)r * N_NODES * HD;
    const int*   srcE = edges_src + (size_t)r * N_EDGES;
    const int*   dstE = edges_dst + (size_t)r * N_EDGES;

    dim3 g1((N_NODES + 63) / 64, HD / 64, 1);
    k_gemm_feat<<<g1, 256, 0, stream>>>(h, W, feat);

    k_attn_logits<<<(int)((nh + 255) / 256), 256, 0, stream>>>(feat, a_all + r * RELW, e_l, e_r);

    hipMemsetAsync(m_enc, 0, nh * sizeof(unsigned), stream);   // 0 <= enc(x) for all x: lower bound
    hipMemsetAsync(s_sum, 0, nh * sizeof(float), stream);

    const int eh = N_EDGES * H_HEADS;
    k_edge_max<<<(eh + 255) / 256, 256, 0, stream>>>(srcE, dstE, e_l, e_r, m_enc);
    k_edge_sum<<<(eh + 255) / 256, 256, 0, stream>>>(srcE, dstE, e_l, e_r, m_enc, s_sum);
    k_edge_scatter<<<N_EDGES / (8 * EPW), 256, 0, stream>>>(srcE, dstE, e_l, e_r,
                                                            m_enc, s_sum, feat, outr);
  }

  const int n4 = R_REL * N_NODES * HD / 4;
  k_relu<<<(n4 + 255) / 256, 256, 0, stream>>>(out_feats, n4);
}